// NeighbourAssignment_49778670960962
// MI455X (gfx1250) — compile-verified
//
#include <hip/hip_runtime.h>
#include <hip/hip_bf16.h>

typedef __attribute__((ext_vector_type(2))) float v2f;
typedef __attribute__((ext_vector_type(4))) float v4f;
typedef __attribute__((ext_vector_type(8))) float v8f;

#define CHN 256      // CS == CT == OUT
#define SZ  4        // SIZE (softmax slots)
// LDS layout: K-pairs interleaved. Row = one even k (holds k, k+1 for 64 cols).
// PSTRIDE % 64 == 32 -> upper half-wave (khalf+1) shifts 32 banks: the wave's
// 64 dword accesses per ds_load_b64 cover all 64 banks, conflict-free.
#define PSTRIDE 160  // floats per pair-row: 128 data + 32 pad

// ---------------------------------------------------------------------------
// Kernel 0: zero the accumulator workspace (M padded, A4, cnt) each launch.
// ---------------------------------------------------------------------------
__global__ void na_zero_kernel(float* __restrict__ ws, long n) {
    long i = (long)blockIdx.x * blockDim.x + threadIdx.x;
    long stride = (long)gridDim.x * blockDim.x;
    for (; i < n; i += stride) ws[i] = 0.0f;
}

// ---------------------------------------------------------------------------
// Kernel 1: per-edge softmax assignment + scatter-accumulate.
//   M[n, s, c]  += a[e,s] * msg[e,c]     (Npad x 4 x 256 fp32, L2-resident)
//   A4[n, s]    += a[e,s]
//   cnt[n]      += 1
// One wave32 per edge; lane l owns channels c = l + 32*j (coalesced).
// Edge data is stream-once -> nontemporal loads keep M hot in L2.
// ---------------------------------------------------------------------------
__global__ __launch_bounds__(256) void na_edge_scatter_kernel(
    const float* __restrict__ own, const float* __restrict__ msg,
    const float* __restrict__ Wt,  const float* __restrict__ bt,
    const float* __restrict__ Ws,  const float* __restrict__ bs,
    const int*   __restrict__ idx,
    float* __restrict__ M, float* __restrict__ A4, float* __restrict__ cnt,
    int E) {
    int wave = threadIdx.x >> 5;
    int lane = threadIdx.x & 31;
    int e = blockIdx.x * 8 + wave;
    if (e >= E) return;

    const float* o = own + (size_t)e * CHN;
    const float* m = msg + (size_t)e * CHN;

    float mv[8];
    float part[SZ] = {0.f, 0.f, 0.f, 0.f};
#pragma unroll
    for (int j = 0; j < 8; ++j) {
        int c = lane + 32 * j;
        float ov = __builtin_nontemporal_load(o + c);
        float mm = __builtin_nontemporal_load(m + c);
        mv[j] = mm;
        v4f wt = *(const v4f*)(Wt + (size_t)c * SZ);   // Wt row c: 4 floats
        v4f ws = *(const v4f*)(Ws + (size_t)c * SZ);
#pragma unroll
        for (int s = 0; s < SZ; ++s) part[s] += ov * wt[s] + mm * ws[s];
    }

    // butterfly reduction across the wave32: all lanes end with full dot
#pragma unroll
    for (int off = 16; off > 0; off >>= 1) {
#pragma unroll
        for (int s = 0; s < SZ; ++s) part[s] += __shfl_xor(part[s], off, 32);
    }

    // stable 4-way softmax (redundant per lane; logits are wave-uniform)
    float lg[SZ], a[SZ];
    float mx = -3.0e38f;
#pragma unroll
    for (int s = 0; s < SZ; ++s) {
        lg[s] = part[s] + bt[s] + bs[s];
        mx = fmaxf(mx, lg[s]);
    }
    float sum = 0.f;
#pragma unroll
    for (int s = 0; s < SZ; ++s) { a[s] = __expf(lg[s] - mx); sum += a[s]; }
    float inv = 1.0f / sum;
#pragma unroll
    for (int s = 0; s < SZ; ++s) a[s] *= inv;

    int n = idx[e];
    float* Mb = M + (size_t)n * (SZ * CHN);
#pragma unroll
    for (int j = 0; j < 8; ++j) {
        int c = lane + 32 * j;
#pragma unroll
        for (int s = 0; s < SZ; ++s)
            atomicAdd(Mb + s * CHN + c, a[s] * mv[j]);   // global_atomic_add_f32
    }
    if (lane == 0) {
        atomicAdd(cnt + n, 1.0f);
#pragma unroll
        for (int s = 0; s < SZ; ++s) atomicAdd(A4 + (size_t)n * SZ + s, a[s]);
    }
}

// ---------------------------------------------------------------------------
// Kernel 2: out[s,n,:] = ( M[n,s,:] @ W_bank[s] + A4[n,s]*b_bank[s,:] )
//                        / max(cnt[n], 1)
// V_WMMA_F32_16X16X4_F32 (fp32 A/B, fp32 acc: exact parity with reference).
// Block = 256 thr (8 waves). blockIdx = (rowTileGroup, colGroup 0..3, s 0..3).
// W slab (256k x 64col) staged in LDS with K-pairs interleaved: each B
// operand {W[k,col], W[k+1,col]} is a single aligned ds_load_b64 into an
// even VGPR pair (no assembly moves), and PSTRIDE=160 makes every b64 read
// bank-conflict-free across the wave.
// Each wave owns one 16-row tile and FOUR 16-col tiles: one 8B A load feeds
// 4 independent WMMAs (distinct D accumulators -> no RAW hazard, b2b issue),
// and cuts global M re-reads 4x. M is padded to Npad rows and zeroed, so the
// A loads need no bounds guard -> no exec-mask churn around loads.
// ---------------------------------------------------------------------------
__global__ __launch_bounds__(256) void na_wmma_gemm_kernel(
    const float* __restrict__ M,   const float* __restrict__ A4,
    const float* __restrict__ cnt, const float* __restrict__ Wb,
    const float* __restrict__ bb,  float* __restrict__ out,
    int N, int Npad) {
    __shared__ float Wtile[(CHN / 2) * PSTRIDE];   // 128 * 160 * 4B = 80 KB

    int s    = blockIdx.z;
    int colG = blockIdx.y;                         // 64-column group

    const float* Wsrc = Wb + (size_t)s * CHN * CHN + (size_t)colG * 64;
    for (int q = threadIdx.x; q < CHN * 64; q += 256) {
        int k = q >> 6, col = q & 63;
        Wtile[(k >> 1) * PSTRIDE + col * 2 + (k & 1)] = Wsrc[(size_t)k * CHN + col];
    }
    __syncthreads();

    int wave = threadIdx.x >> 5;
    int lane = threadIdx.x & 31;
    int NT   = Npad >> 4;
    int rowT = blockIdx.x * 8 + wave;
    if (rowT >= NT) return;                        // uniform per wave: EXEC full

    int rl    = lane & 15;                         // row (A) / col (B) in tile
    int kh    = (lane >> 4) << 1;                  // K sub-offset: 0 or 2
    int khalf = lane >> 4;                         // pair-row sub-offset: 0 or 1
    int r     = rowT * 16 + rl;                    // < Npad: M padded + zeroed
    const float* Mrow = M + (size_t)r * (SZ * CHN) + (size_t)s * CHN + kh;

    v8f c0 = {0.f,0.f,0.f,0.f,0.f,0.f,0.f,0.f};
    v8f c1 = c0, c2 = c0, c3 = c0;

#pragma unroll 8
    for (int k0 = 0; k0 < CHN; k0 += 4) {
        v2f a = *(const v2f*)(Mrow + k0);          // A[r, k0+kh .. +1]
        const float* bp = Wtile + ((k0 >> 1) + khalf) * PSTRIDE + rl * 2;
        v2f b0 = *(const v2f*)(bp +  0);           // cols  0..15 (pair k,k+1)
        v2f b1 = *(const v2f*)(bp + 32);           // cols 16..31
        v2f b2 = *(const v2f*)(bp + 64);           // cols 32..47
        v2f b3 = *(const v2f*)(bp + 96);           // cols 48..63
        c0 = __builtin_amdgcn_wmma_f32_16x16x4_f32(false, a, false, b0, (short)0, c0, false, false);
        c1 = __builtin_amdgcn_wmma_f32_16x16x4_f32(false, a, false, b1, (short)0, c1, false, false);
        c2 = __builtin_amdgcn_wmma_f32_16x16x4_f32(false, a, false, b2, (short)0, c2, false, false);
        c3 = __builtin_amdgcn_wmma_f32_16x16x4_f32(false, a, false, b3, (short)0, c3, false, false);
    }

    // C/D layout: VGPR i -> row i (lanes 0-15) / row i+8 (lanes 16-31)
    int rowHi = (lane >> 4) * 8;
    v8f acc[4] = { c0, c1, c2, c3 };
#pragma unroll
    for (int ct = 0; ct < 4; ++ct) {
        int ocol = colG * 64 + ct * 16 + rl;
        float bv = bb[(size_t)s * CHN + ocol];
#pragma unroll
        for (int i = 0; i < 8; ++i) {
            int n = rowT * 16 + i + rowHi;
            if (n < N) {
                float denom = fmaxf(cnt[n], 1.0f);
                float av = A4[(size_t)n * SZ + s];
                __builtin_nontemporal_store(
                    (acc[ct][i] + av * bv) / denom,
                    out + ((size_t)s * N + n) * CHN + ocol);
            }
        }
    }
}

// ---------------------------------------------------------------------------
// Launch
// ---------------------------------------------------------------------------
extern "C" void kernel_launch(void* const* d_in, const int* in_sizes, int n_in,
                              void* d_out, int out_size, void* d_ws, size_t ws_size,
                              hipStream_t stream) {
    const float* own = (const float*)d_in[0];
    const float* msg = (const float*)d_in[1];
    const float* Wt  = (const float*)d_in[2];
    const float* bt  = (const float*)d_in[3];
    const float* Ws  = (const float*)d_in[4];
    const float* bs  = (const float*)d_in[5];
    const float* Wb  = (const float*)d_in[6];
    const float* bb  = (const float*)d_in[7];
    const int*   idx = (const int*)d_in[8];
    float*       out = (float*)d_out;

    int E    = in_sizes[0] / CHN;             // 160000
    int N    = out_size / (SZ * CHN);         // 10000
    int NT   = (N + 15) / 16;
    int Npad = NT * 16;

    // workspace: M (Npad*4*256) | A4 (N*4) | cnt (N)   -- all zeroed per call
    float* M   = (float*)d_ws;
    float* A4  = M + (size_t)Npad * (SZ * CHN);
    float* cnt = A4 + (size_t)N * SZ;
    long   zn  = (long)Npad * (SZ * CHN) + (long)N * SZ + N;

    na_zero_kernel<<<2048, 256, 0, stream>>>((float*)d_ws, zn);

    na_edge_scatter_kernel<<<(E + 7) / 8, 256, 0, stream>>>(
        own, msg, Wt, bt, Ws, bs, idx, M, A4, cnt, E);

    dim3 grid((NT + 7) / 8, 4, SZ);
    na_wmma_gemm_kernel<<<grid, 256, 0, stream>>>(M, A4, cnt, Wb, bb, out, N, Npad);
}